// EMLMessagePassing_3332894621797
// MI455X (gfx1250) — compile-verified
//
#include <hip/hip_runtime.h>
#include <hip/hip_bf16.h>
#include <math.h>

typedef __attribute__((ext_vector_type(2))) float v2f;
typedef __attribute__((ext_vector_type(8))) float v8f;

// Problem dims
// B=8 batches, N=128 slots, D=64, E=128, H=256, 6D=384

// ---------------- workspace layout (float offsets) ----------------
constexpr int OFF_EV   = 0;                    // 8*64
constexpr int OFF_TGT  = OFF_EV   + 8*64;      // 8*128*64
constexpr int OFF_SRC  = OFF_TGT  + 65536;
constexpr int OFF_VALS = OFF_SRC  + 65536;
constexpr int OFF_WDT  = OFF_VALS + 65536;     // 64*256 (drive, T coeff)
constexpr int OFF_WDS  = OFF_WDT  + 16384;     // 64*256 (drive, S coeff)
constexpr int OFF_WRT  = OFF_WDS  + 16384;
constexpr int OFF_WRS  = OFF_WRT  + 16384;
constexpr int OFF_SGWD = OFF_WRS  + 16384;     // 256  sum(g*W1row) drive
constexpr int OFF_LBWD = OFF_SGWD + 256;       // 256  ln_b·W1row + b1 drive
constexpr int OFF_SGWR = OFF_LBWD + 256;
constexpr int OFF_LBWR = OFF_SGWR + 256;
constexpr int OFF_CD   = OFF_LBWR + 256;       // 8*256 event proj drive
constexpr int OFF_CR   = OFF_CD   + 2048;
constexpr int OFF_AD   = OFF_CR   + 2048;      // 8*128*256 tgt proj drive
constexpr int OFF_BD   = OFF_AD   + 262144;    // src proj drive
constexpr int OFF_AR   = OFF_BD   + 262144;
constexpr int OFF_BR   = OFF_AR   + 262144;
constexpr int OFF_TS   = OFF_BR   + 262144;    // 8*128*128  tgt_i . src_j
constexpr int OFF_GATE = OFF_TS   + 131072;    // 8*128*128
constexpr int OFF_MASS = OFF_GATE + 131072;    // 8*128
constexpr int OFF_SUMS = OFF_MASS + 1024;      // 8*128 sum(src_j)
constexpr int OFF_NRMS = OFF_SUMS + 1024;      // 8*128 |src_j|^2
constexpr int OFF_NRMT = OFF_NRMS + 1024;      // 8*128 |tgt_i|^2
constexpr int OFF_SEV  = OFF_NRMT + 1024;      // 8 sum(ev)
constexpr int OFF_NEV  = OFF_SEV  + 8;         // 8 |ev|^2

__device__ inline float gelu_exact(float x) {
  return 0.5f * x * (1.0f + erff(x * 0.70710678118654752440f));
}

// ---------------- stage 1: tiny precompute (one block) ----------------
__global__ __launch_bounds__(256) void prep_kernel(
    const float* __restrict__ event, const float* __restrict__ We, const float* __restrict__ be,
    const float* __restrict__ ln_g, const float* __restrict__ ln_b,
    const float* __restrict__ dW1, const float* __restrict__ db1,
    const float* __restrict__ rW1, const float* __restrict__ rb1,
    float* __restrict__ ws) {
  const int tid = threadIdx.x;
  float* ev = ws + OFF_EV;
  // ev[b][d] = event[b,:] . We[d,:] + be[d]
  for (int o = tid; o < 8 * 64; o += 256) {
    int b = o >> 6, d = o & 63;
    float acc = be[d];
    for (int e = 0; e < 128; ++e) acc = fmaf(event[b * 128 + e], We[d * 128 + e], acc);
    ev[o] = acc;
  }
  // Fold LN gain into W1 blocks: T coeff = g0*W1_blk0 - g3*W1_blk3; S coeff = g1*W1_blk1 + g3*W1_blk3
  float* WdT = ws + OFF_WDT; float* WdS = ws + OFF_WDS;
  float* WrT = ws + OFF_WRT; float* WrS = ws + OFF_WRS;
  for (int o = tid; o < 64 * 256; o += 256) {
    int d = o >> 8, k = o & 255;
    float g0 = ln_g[d], g1 = ln_g[64 + d], g3 = ln_g[192 + d];
    float w0 = dW1[k * 384 + d], w1 = dW1[k * 384 + 64 + d], w3 = dW1[k * 384 + 192 + d];
    WdT[o] = g0 * w0 - g3 * w3;
    WdS[o] = g1 * w1 + g3 * w3;
    float r0 = rW1[k * 384 + d], r1 = rW1[k * 384 + 64 + d], r3 = rW1[k * 384 + 192 + d];
    WrT[o] = g0 * r0 - g3 * r3;
    WrS[o] = g1 * r1 + g3 * r3;
  }
  // Per-hidden-unit LN constants
  if (tid < 256) {
    int k = tid;
    float sgd = 0.f, lbd = 0.f, sgr = 0.f, lbr = 0.f;
    for (int m = 0; m < 384; ++m) {
      float g = ln_g[m], lb = ln_b[m];
      float wd = dW1[k * 384 + m], wr = rW1[k * 384 + m];
      sgd = fmaf(g, wd, sgd);  lbd = fmaf(lb, wd, lbd);
      sgr = fmaf(g, wr, sgr);  lbr = fmaf(lb, wr, lbr);
    }
    ws[OFF_SGWD + k] = sgd; ws[OFF_LBWD + k] = lbd + db1[k];
    ws[OFF_SGWR + k] = sgr; ws[OFF_LBWR + k] = lbr + rb1[k];
  }
  __syncthreads();
  // Event projections C[b][k] and EV stats
  for (int o = tid; o < 8 * 256; o += 256) {
    int b = o >> 8, k = o & 255;
    float cd = 0.f, cr = 0.f;
    for (int d = 0; d < 64; ++d) {
      float evd = ev[b * 64 + d] * ln_g[128 + d];
      cd = fmaf(evd, dW1[k * 384 + 128 + d], cd);
      cr = fmaf(evd, rW1[k * 384 + 128 + d], cr);
    }
    ws[OFF_CD + o] = cd; ws[OFF_CR + o] = cr;
  }
  if (tid < 8) {
    float s = 0.f, n = 0.f;
    for (int d = 0; d < 64; ++d) { float v = ev[tid * 64 + d]; s += v; n = fmaf(v, v, n); }
    ws[OFF_SEV + tid] = s; ws[OFF_NEV + tid] = n;
  }
}

// ---------------- stage 2: slot transforms + row stats ----------------
__global__ __launch_bounds__(64) void xform_kernel(
    const float* __restrict__ x,
    const float* __restrict__ Wsrc, const float* __restrict__ bsrc,
    const float* __restrict__ Wtgt, const float* __restrict__ btgt,
    const float* __restrict__ Wval, const float* __restrict__ bval,
    float* __restrict__ ws) {
  __shared__ float xr[64];
  __shared__ float r0[64], r1[64], r2[64];
  const int bn = blockIdx.x;           // b*128 + n
  const int d = threadIdx.x;
  xr[d] = x[bn * 64 + d];
  __syncthreads();
  float s = bsrc[d], t = btgt[d], v = bval[d];
  for (int c = 0; c < 64; ++c) {
    float xv = xr[c];
    s = fmaf(Wsrc[d * 64 + c], xv, s);
    t = fmaf(Wtgt[d * 64 + c], xv, t);
    v = fmaf(Wval[d * 64 + c], xv, v);
  }
  ws[OFF_SRC  + bn * 64 + d] = s;
  ws[OFF_TGT  + bn * 64 + d] = t;
  ws[OFF_VALS + bn * 64 + d] = v;
  r0[d] = s; r1[d] = s * s; r2[d] = t * t;
  __syncthreads();
  for (int off = 32; off > 0; off >>= 1) {
    if (d < off) { r0[d] += r0[d + off]; r1[d] += r1[d + off]; r2[d] += r2[d + off]; }
    __syncthreads();
  }
  if (d == 0) {
    ws[OFF_SUMS + bn] = r0[0];
    ws[OFF_NRMS + bn] = r1[0];
    ws[OFF_NRMT + bn] = r2[0];
  }
}

// ---------------- WMMA helper: 16x16 f32 tile, row-major A[lda], B[ldb] ----------------
__device__ inline v8f wmma16x16_f32(const float* __restrict__ A, int lda,
                                    const float* __restrict__ Bp, int ldb, int K) {
  const int lane = threadIdx.x & 31;
  const int m = lane & 15;
  const int h = lane >> 4;             // lane half selects K pair
  v8f acc = {};
  for (int kb = 0; kb < K; kb += 4) {
    int k0 = kb + 2 * h;
    v2f a, b;
    a.x = A[m * lda + k0];
    a.y = A[m * lda + k0 + 1];
    b.x = Bp[k0 * ldb + m];
    b.y = Bp[(k0 + 1) * ldb + m];
    acc = __builtin_amdgcn_wmma_f32_16x16x4_f32(false, a, false, b, (short)0, acc, false, false);
  }
  return acc;
}

// ---------------- stage 3: hidden-unit projections A/B (per batch GEMMs) ----------------
__global__ __launch_bounds__(32) void proj_wmma_kernel(float* __restrict__ ws) {
  const int it = blockIdx.x;           // 0..7 (rows of 16)
  const int nt = blockIdx.y;           // 0..15 (cols of 16)
  const int z  = blockIdx.z;           // b*4 + which
  const int b = z >> 2, which = z & 3;
  const float* A  = ws + ((which & 1) == 0 ? OFF_TGT : OFF_SRC) + b * 128 * 64;
  const float* Bp = ws + (which == 0 ? OFF_WDT : which == 1 ? OFF_WDS : which == 2 ? OFF_WRT : OFF_WRS);
  float*       C  = ws + (which == 0 ? OFF_AD  : which == 1 ? OFF_BD  : which == 2 ? OFF_AR  : OFF_BR) + b * 128 * 256;
  v8f acc = wmma16x16_f32(A + it * 16 * 64, 64, Bp + nt * 16, 256, 64);
  const int lane = threadIdx.x & 31;
  const int m = lane & 15, h = lane >> 4;
#pragma unroll
  for (int r = 0; r < 8; ++r)
    C[(it * 16 + r + 8 * h) * 256 + nt * 16 + m] = acc[r];
}

// ---------------- stage 4: TS[b][i][j] = tgt_i . src_j ----------------
__global__ __launch_bounds__(32) void ts_wmma_kernel(float* __restrict__ ws) {
  const int it = blockIdx.x, jt = blockIdx.y, b = blockIdx.z;
  const float* T = ws + OFF_TGT + b * 8192 + it * 16 * 64;
  const float* S = ws + OFF_SRC + b * 8192 + jt * 16 * 64;  // B = S^T -> B[k][n] = S[n][k]
  const int lane = threadIdx.x & 31;
  const int m = lane & 15, h = lane >> 4;
  v8f acc = {};
  for (int kb = 0; kb < 64; kb += 4) {
    int k0 = kb + 2 * h;
    v2f a, bb;
    a.x  = T[m * 64 + k0]; a.y  = T[m * 64 + k0 + 1];
    bb.x = S[m * 64 + k0]; bb.y = S[m * 64 + k0 + 1];
    acc = __builtin_amdgcn_wmma_f32_16x16x4_f32(false, a, false, bb, (short)0, acc, false, false);
  }
  float* TS = ws + OFF_TS + b * 16384;
#pragma unroll
  for (int r = 0; r < 8; ++r)
    TS[(it * 16 + r + 8 * h) * 128 + jt * 16 + m] = acc[r];
}

// ---------------- stage 5: gate hot loop (one wave32 per (b,i)) ----------------
__global__ __launch_bounds__(32) void gate_kernel(
    float* __restrict__ ws,
    const float* __restrict__ dW2, const float* __restrict__ db2,
    const float* __restrict__ rW2, const float* __restrict__ rb2,
    const float* __restrict__ gate_bias) {
  const int bi = blockIdx.x;           // b*128 + i
  const int b = bi >> 7, i = bi & 127;
  const int lane = threadIdx.x & 31;
  float ad[8], ar[8], cdv[8], crv[8], sgd[8], lbd[8], sgr[8], lbr[8], w2d[8], w2r[8];
#pragma unroll
  for (int t = 0; t < 8; ++t) {
    int k = lane + 32 * t;
    ad[t]  = ws[OFF_AD + bi * 256 + k];
    ar[t]  = ws[OFF_AR + bi * 256 + k];
    cdv[t] = ws[OFF_CD + b * 256 + k];
    crv[t] = ws[OFF_CR + b * 256 + k];
    sgd[t] = ws[OFF_SGWD + k];  lbd[t] = ws[OFF_LBWD + k];
    sgr[t] = ws[OFF_SGWR + k];  lbr[t] = ws[OFF_LBWR + k];
    w2d[t] = dW2[k];            w2r[t] = rW2[k];
  }
  const float nT  = ws[OFF_NRMT + bi];
  const float nEV = ws[OFF_NEV + b];
  const float sEV = ws[OFF_SEV + b];
  const float db2v = db2[0], rb2v = rb2[0], gb = gate_bias[0];
  float mass = 0.f;
  const float* Bd = ws + OFF_BD + b * 128 * 256;
  const float* Br = ws + OFF_BR + b * 128 * 256;
  for (int j = 0; j < 128; ++j) {
    float sS = ws[OFF_SUMS + b * 128 + j];
    float nS = ws[OFF_NRMS + b * 128 + j];
    float ts = ws[OFF_TS + bi * 128 + j];
    float mu = (2.f * sS + sEV) * (1.f / 384.f);
    float sumsq = 2.f * nT + 2.f * nS + nEV - 2.f * ts;
    float var = sumsq * (1.f / 384.f) - mu * mu;
    float rsv = rsqrtf(var + 1e-5f);
    const float* bd = Bd + j * 256;
    const float* br = Br + j * 256;
    float dsum = 0.f, rsum = 0.f;
#pragma unroll
    for (int t = 0; t < 8; ++t) {
      int k = lane + 32 * t;
      float hd = fmaf(rsv, ad[t] + bd[k] + cdv[t] - mu * sgd[t], lbd[t]);
      dsum = fmaf(w2d[t], gelu_exact(hd), dsum);
      float hr = fmaf(rsv, ar[t] + br[k] + crv[t] - mu * sgr[t], lbr[t]);
      rsum = fmaf(w2r[t], gelu_exact(hr), rsum);
    }
#pragma unroll
    for (int off = 16; off > 0; off >>= 1) {
      dsum += __shfl_xor(dsum, off, 32);
      rsum += __shfl_xor(rsum, off, 32);
    }
    float drive = dsum + db2v;
    float resv  = rsum + rb2v;
    float sp = (resv > 20.f) ? resv : log1pf(expf(resv));
    float energy = drive / (sp + 1e-6f);
    energy = fminf(fmaxf(energy, -3.f), 3.f);
    float g = 1.f / (1.f + expf(-(energy + gb)));
    if (j == i) g = 0.f;
    mass += g;
    if (lane == 0) ws[OFF_GATE + bi * 128 + j] = g;
  }
  if (lane == 0) ws[OFF_MASS + bi] = fmaxf(mass, 1e-6f);
}

// ---------------- stage 6: out = (gate @ values) / mass ----------------
__global__ __launch_bounds__(32) void agg_wmma_kernel(float* __restrict__ ws, float* __restrict__ out) {
  const int it = blockIdx.x, dt = blockIdx.y, b = blockIdx.z;
  const float* G = ws + OFF_GATE + b * 16384 + it * 16 * 128;
  const float* V = ws + OFF_VALS + b * 8192 + dt * 16;
  v8f acc = wmma16x16_f32(G, 128, V, 64, 128);
  const int lane = threadIdx.x & 31;
  const int m = lane & 15, h = lane >> 4;
#pragma unroll
  for (int r = 0; r < 8; ++r) {
    int row = it * 16 + r + 8 * h;
    float mass = ws[OFF_MASS + b * 128 + row];
    out[(b * 128 + row) * 64 + dt * 16 + m] = acc[r] / mass;
  }
}

extern "C" void kernel_launch(void* const* d_in, const int* in_sizes, int n_in,
                              void* d_out, int out_size, void* d_ws, size_t ws_size,
                              hipStream_t stream) {
  const float* x       = (const float*)d_in[0];
  const float* event_  = (const float*)d_in[1];
  const float* We      = (const float*)d_in[2];
  const float* be      = (const float*)d_in[3];
  const float* Wsrc    = (const float*)d_in[4];
  const float* bsrc    = (const float*)d_in[5];
  const float* Wtgt    = (const float*)d_in[6];
  const float* btgt    = (const float*)d_in[7];
  const float* Wval    = (const float*)d_in[8];
  const float* bval    = (const float*)d_in[9];
  const float* ln_g    = (const float*)d_in[10];
  const float* ln_b    = (const float*)d_in[11];
  const float* dW1     = (const float*)d_in[12];
  const float* db1     = (const float*)d_in[13];
  const float* dW2     = (const float*)d_in[14];
  const float* db2     = (const float*)d_in[15];
  const float* rW1     = (const float*)d_in[16];
  const float* rb1     = (const float*)d_in[17];
  const float* rW2     = (const float*)d_in[18];
  const float* rb2     = (const float*)d_in[19];
  const float* gbias   = (const float*)d_in[20];
  float* ws  = (float*)d_ws;
  float* out = (float*)d_out;

  hipLaunchKernelGGL(prep_kernel, dim3(1), dim3(256), 0, stream,
                     event_, We, be, ln_g, ln_b, dW1, db1, rW1, rb1, ws);
  hipLaunchKernelGGL(xform_kernel, dim3(1024), dim3(64), 0, stream,
                     x, Wsrc, bsrc, Wtgt, btgt, Wval, bval, ws);
  hipLaunchKernelGGL(proj_wmma_kernel, dim3(8, 16, 32), dim3(32), 0, stream, ws);
  hipLaunchKernelGGL(ts_wmma_kernel, dim3(8, 8, 8), dim3(32), 0, stream, ws);
  hipLaunchKernelGGL(gate_kernel, dim3(1024), dim3(32), 0, stream,
                     ws, dW2, db2, rW2, rb2, gbias);
  hipLaunchKernelGGL(agg_wmma_kernel, dim3(8, 4, 8), dim3(32), 0, stream, ws, out);
}